// GaussianKernelGCNLayer_89361089561157
// MI455X (gfx1250) — compile-verified
//
#include <hip/hip_runtime.h>

// MI455X / gfx1250, wave32. fp32 WMMA 16x16x4 GEMM + async global->LDS staging (inline asm).

typedef __attribute__((ext_vector_type(2))) float v2f;
typedef __attribute__((ext_vector_type(8))) float v8f;

#if defined(__gfx1250__)
#define HAS_ASYNC_LDS 1
#else
#define HAS_ASYNC_LDS 0
#endif

#define NBATCH   2048      // NB*NI
#define NN       32
#define DIN      4096
#define NK       8
#define KOUT     512       // DOUT / NK
#define MT       64        // block M tile (b)
#define NT       128       // block N tile (o)
#define KT       16        // K step (d)
#define LDA      20        // A LDS stride: 16B-aligned float4 slots AND 20*l mod 64 distinct -> conflict-free
#define LDB      136       // B LDS stride: 2*136 mod 64 = 16 -> lane halves hit disjoint banks; 16B aligned

// 16-byte global -> LDS copy: async LDS-DMA on gfx1250, else through VGPRs.
// LDS address = low 32 bits of the flat pointer (ISA 10.2).
__device__ __forceinline__ void copy16(const float* __restrict__ g, float* l) {
#if HAS_ASYNC_LDS
    asm volatile("global_load_async_to_lds_b128 %0, %1, off"
                 :
                 : "v"((unsigned int)(unsigned long long)l),
                   "v"((unsigned long long)g)
                 : "memory");
#else
    float4 v = *(const float4*)g;
    l[0] = v.x; l[1] = v.y; l[2] = v.z; l[3] = v.w;
#endif
}

__device__ __forceinline__ void wait_async() {
#if HAS_ASYNC_LDS
    asm volatile("s_wait_asynccnt 0x0" ::: "memory");
#endif
}

// ---------------- Kernel A: wf[b,k,d] = sum_n w[b,n,k] * f[b,n,d] ----------------
// 256 threads, 4 d's/thread (b128 stream of the dominant 1.07GB feats read).
__global__ __launch_bounds__(256) void agg_kernel(const float* __restrict__ f,
                                                  const float* __restrict__ w,
                                                  float* __restrict__ wf,
                                                  int b0) {
    __shared__ float sw[NN * NK];
    const int t    = threadIdx.x;
    const int segs = DIN / 1024;             // 4 blocks per b
    const int bl   = blockIdx.x / segs;
    const int b    = b0 + bl;
    const int d    = (blockIdx.x % segs) * 1024 + t * 4;

    sw[t] = w[(size_t)b * (NN * NK) + t];    // w[b][n][k] at n*8+k
    __syncthreads();

    float4 acc[NK] = {};
    const float* fb = f + (size_t)b * NN * DIN + d;
#pragma unroll 4
    for (int n = 0; n < NN; ++n) {
        float4 fv = *(const float4*)(fb + (size_t)n * DIN);   // b128 coalesced
#pragma unroll
        for (int k = 0; k < NK; ++k) {
            float s = sw[n * NK + k];                          // LDS broadcast
            acc[k].x = fmaf(s, fv.x, acc[k].x);
            acc[k].y = fmaf(s, fv.y, acc[k].y);
            acc[k].z = fmaf(s, fv.z, acc[k].z);
            acc[k].w = fmaf(s, fv.w, acc[k].w);
        }
    }
    float* dst = wf + (size_t)bl * NK * DIN + d;               // layout [bl][k][DIN]
#pragma unroll
    for (int k = 0; k < NK; ++k)
        *(float4*)(dst + (size_t)k * DIN) = acc[k];            // b128 coalesced
}

// ---------------- Kernel B: out[b, k*512+o] = wf[b,k,:] @ cw[k,:,:] ----------------
// 8 waves, block tile 64x128, wave tile 16x64 (4 C tiles), double-buffered async LDS.
__global__ __launch_bounds__(256) void gemm_kernel(const float* __restrict__ wf,
                                                   const float* __restrict__ cw,
                                                   float* __restrict__ out,
                                                   int b0, int mtiles) {
    __shared__ float sA[2][MT * LDA];        // 2 x 64x16 (stride 20)  = 10.0 KB
    __shared__ float sB[2][KT * LDB];        // 2 x 16x128 (stride 136)= 17.0 KB

    const int gid   = blockIdx.x;
    const int ntile = gid % (KOUT / NT);                 // 0..3
    const int mt    = (gid / (KOUT / NT)) % mtiles;      // b tile
    const int k     = gid / ((KOUT / NT) * mtiles);      // kernel index 0..7

    const int t    = threadIdx.x;
    const int wave = t >> 5;
    const int lane = t & 31;
    const int l16  = lane & 15;
    const int lh   = lane >> 4;              // lane half -> K pair / M half
    const int wm   = wave >> 1;              // 0..3 -> M subtile
    const int wn   = wave & 1;               // 0..1 -> N half (64 cols)

    const float* Abase = wf + ((size_t)(mt * MT) * NK + k) * DIN;       // row stride NK*DIN
    const float* Bbase = cw + (size_t)k * DIN * KOUT + (size_t)ntile * NT;

    // staging coordinates (one b128 for A, two b128 for B per thread per tile)
    const int ra = t >> 2, ca = (t & 3) << 2;            // A: 64 rows x 4 threads
    const int rb = t >> 4, cb = (t & 15) << 3;           // B: 16 rows x 16 threads x 8 floats
    const float* gA = Abase + (size_t)ra * (NK * DIN) + ca;
    const float* gB = Bbase + (size_t)rb * KOUT + cb;

    auto stage = [&](int buf, int kb) {
        copy16(gA + kb,                     &sA[buf][ra * LDA + ca]);
        copy16(gB + (size_t)kb * KOUT,      &sB[buf][rb * LDB + cb]);
        copy16(gB + (size_t)kb * KOUT + 4,  &sB[buf][rb * LDB + cb + 4]);
    };

    v8f c0 = {}, c1 = {}, c2 = {}, c3 = {};

    stage(0, 0);
    int buf = 0;
    for (int kb = 0; kb < DIN; kb += KT, buf ^= 1) {
        wait_async();                 // our loads for 'buf' complete
        __syncthreads();              // everyone's loads done; prev compute on buf^1 done
        if (kb + KT < DIN) stage(buf ^ 1, kb + KT);   // DMA next tile under the math
#pragma unroll
        for (int kk = 0; kk < KT; kk += 4) {
            // A frag (16x4): lanes 0-15 -> K={kk,kk+1}, lanes 16-31 -> K={kk+2,kk+3}
            v2f a;
            {
                const float* ap = &sA[buf][(wm * 16 + l16) * LDA + kk + 2 * lh];
                a.x = ap[0]; a.y = ap[1];
            }
            // B frags (4x16): v0 = K row (kk+2*lh), v1 = next K row
            const float* bp = &sB[buf][(kk + 2 * lh) * LDB + wn * 64 + l16];
            v2f bb0; bb0.x = bp[0];  bb0.y = bp[LDB];
            v2f bb1; bb1.x = bp[16]; bb1.y = bp[LDB + 16];
            v2f bb2; bb2.x = bp[32]; bb2.y = bp[LDB + 32];
            v2f bb3; bb3.x = bp[48]; bb3.y = bp[LDB + 48];
            c0 = __builtin_amdgcn_wmma_f32_16x16x4_f32(false, a, false, bb0, (short)0, c0, false, false);
            c1 = __builtin_amdgcn_wmma_f32_16x16x4_f32(false, a, false, bb1, (short)0, c1, false, false);
            c2 = __builtin_amdgcn_wmma_f32_16x16x4_f32(false, a, false, bb2, (short)0, c2, false, false);
            c3 = __builtin_amdgcn_wmma_f32_16x16x4_f32(false, a, false, bb3, (short)0, c3, false, false);
        }
    }

    // C store: VGPR i holds M=i (lanes 0-15) / M=i+8 (lanes 16-31), N=l16
    const size_t rowBase = (size_t)(b0 + mt * MT + wm * 16);
    const int    oBase   = k * KOUT + ntile * NT + wn * 64 + l16;
#pragma unroll
    for (int i = 0; i < 8; ++i) {
        size_t idx = (rowBase + i + 8 * lh) * (size_t)(NK * KOUT) + oBase;
        out[idx]      = c0[i];
        out[idx + 16] = c1[i];
        out[idx + 32] = c2[i];
        out[idx + 48] = c3[i];
    }
}

extern "C" void kernel_launch(void* const* d_in, const int* in_sizes, int n_in,
                              void* d_out, int out_size, void* d_ws, size_t ws_size,
                              hipStream_t stream) {
    (void)in_sizes; (void)n_in; (void)out_size;
    const float* feats = (const float*)d_in[0];   // [8,256,32,4096] f32
    const float* wgt   = (const float*)d_in[1];   // [8,256,32,8]    f32
    const float* cw    = (const float*)d_in[2];   // [8,4096,512]    f32
    float*       out   = (float*)d_out;           // [2048, 4096]    f32
    float*       wf    = (float*)d_ws;            // chunk workspace [bc][8][4096] f32

    const size_t perB = (size_t)NK * DIN * sizeof(float);   // 128 KB per instance
    int bc = NBATCH;
    while ((size_t)bc * perB > ws_size && bc > MT) bc >>= 1;

    for (int b0 = 0; b0 < NBATCH; b0 += bc) {
        agg_kernel<<<bc * (DIN / 1024), 256, 0, stream>>>(feats, wgt, wf, b0);
        const int mtiles = bc / MT;
        gemm_kernel<<<NK * mtiles * (KOUT / NT), 256, 0, stream>>>(wf, cw, out, b0, mtiles);
    }
}